// TwistJMotorInt8_80530636800249
// MI455X (gfx1250) — compile-verified
//
#include <hip/hip_runtime.h>
#include <stdint.h>
#include <math.h>

#define DIM   4096
#define TPB   256            // 8 wave32 per block; 1 block == 1 row
#define EPT   16             // elements per thread (TPB*EPT == DIM)
#define NW    (TPB / 32)

typedef float v4f __attribute__((ext_vector_type(4)));
typedef int   v4i __attribute__((vector_size(16)));   // matches builtin's V4i
typedef __attribute__((address_space(1))) v4i* gv4i_p;
typedef __attribute__((address_space(3))) v4i* sv4i_p;

// ---- CDNA5 async global->LDS path (probe-guarded) -------------------------
#if defined(__has_builtin)
#  if __has_builtin(__builtin_amdgcn_global_load_async_to_lds_b128)
#    define USE_ASYNC_LDS 1
#  endif
#  if __has_builtin(__builtin_amdgcn_s_wait_asynccnt)
#    define HAVE_WAIT_ASYNCCNT 1
#  endif
#endif
#ifndef USE_ASYNC_LDS
#  define USE_ASYNC_LDS 0
#endif
#ifndef HAVE_WAIT_ASYNCCNT
#  define HAVE_WAIT_ASYNCCNT 0
#endif

__global__ __launch_bounds__(TPB)
void twist_mj_rule42_kernel(const float* __restrict__ x,
                            const float* __restrict__ scale,
                            const float* __restrict__ bias,
                            float* __restrict__ out)
{
    __shared__ float s_row[DIM];          // async-DMA landing zone (16 KB)
    __shared__ float s_wmax[NW];          // per-wave row maxima
    __shared__ float s_edge[TPB][4];      // q[0],q[1],q[14],q[15] per thread

    const int t    = threadIdx.x;
    const int lane = t & 31;              // wave32 on gfx1250
    const int wid  = t >> 5;
    const long long base = (long long)blockIdx.x * DIM + (long long)t * EPT;
    const float* gx = x + base;

    float f[EPT];

#if USE_ASYNC_LDS
    {
        // Stage this thread's 64 B via the CDNA5 async data mover:
        // 4x global_load_async_to_lds_b128, tracked on ASYNCcnt.
        float* lb = &s_row[t * EPT];
#pragma unroll
        for (int k = 0; k < 4; ++k) {
            __builtin_amdgcn_global_load_async_to_lds_b128(
                (gv4i_p)(gx + 4 * k),
                (sv4i_p)(lb + 4 * k),
                0, 0);
        }
#if HAVE_WAIT_ASYNCCNT
        __builtin_amdgcn_s_wait_asynccnt(0);
#else
        asm volatile("s_wait_asynccnt 0" ::: "memory");
#endif
        // Each thread reads back exactly the region its own wave DMA'd.
#pragma unroll
        for (int k = 0; k < 4; ++k) {
            v4f v = *(v4f*)(lb + 4 * k);
            f[4 * k + 0] = v[0]; f[4 * k + 1] = v[1];
            f[4 * k + 2] = v[2]; f[4 * k + 3] = v[3];
        }
    }
#else
#pragma unroll
    for (int k = 0; k < 4; ++k) {
        v4f v = *(const v4f*)(gx + 4 * k);
        f[4 * k + 0] = v[0]; f[4 * k + 1] = v[1];
        f[4 * k + 2] = v[2]; f[4 * k + 3] = v[3];
    }
#endif

    float q[EPT];

    // Thue-Morse over DEPTH=4 steps: roll on steps 1 and 2 only.
#pragma unroll
    for (int step = 0; step < 4; ++step) {
        const bool rolled = (step == 1) || (step == 2);

        // ---- row-wise max(|x|) : local -> wave32 shfl tree -> LDS ----
        float m = 0.0f;
#pragma unroll
        for (int i = 0; i < EPT; ++i) m = fmaxf(m, fabsf(f[i]));
#pragma unroll
        for (int off = 16; off > 0; off >>= 1)
            m = fmaxf(m, __shfl_xor(m, off, 32));
        if (lane == 0) s_wmax[wid] = m;
        __syncthreads();
        float rmax = s_wmax[0];
#pragma unroll
        for (int w = 1; w < NW; ++w) rmax = fmaxf(rmax, s_wmax[w]);
        __syncthreads();                       // allow s_wmax reuse next step

        rmax = fmaxf(rmax, 1e-12f);
        const float qs  = rmax * (1.0f / 42.0f);   // q_scale
        const float inv = 1.0f / qs;

        // ---- quantize (exact small ints kept in f32: |q| <= 42) ----
#pragma unroll
        for (int i = 0; i < EPT; ++i)
            q[i] = fminf(fmaxf(rintf(f[i] * inv), -42.0f), 42.0f);

        if (rolled) {
            // need x[-2],x[-1] from left thread and x[16],x[17] from right
            s_edge[t][0] = q[0];  s_edge[t][1] = q[1];
            s_edge[t][2] = q[14]; s_edge[t][3] = q[15];
            __syncthreads();
            const int lt = (t + TPB - 1) & (TPB - 1);  // wrap within row
            const int rt = (t + 1) & (TPB - 1);
            float e[EPT + 4];                          // e[i] == x[i-2]
            e[0] = s_edge[lt][2];
            e[1] = s_edge[lt][3];
#pragma unroll
            for (int i = 0; i < EPT; ++i) e[i + 2] = q[i];
            e[EPT + 2] = s_edge[rt][0];
            e[EPT + 3] = s_edge[rt][1];
            // y[b]=x[b-2]; y[b+1]=x[b-1]-x[b]+x[b+1];
            // y[b+2]=x[b+2]-x[b+4]+x[b+5]; y[b+3]=x[b+3]-x[b+4]
#pragma unroll
            for (int b = 0; b < EPT; b += 4) {
                f[b + 0] = e[b + 0] * qs;
                f[b + 1] = (e[b + 1] - e[b + 2] + e[b + 3]) * qs;
                f[b + 2] = (e[b + 4] - e[b + 6] + e[b + 7]) * qs;
                f[b + 3] = (e[b + 5] - e[b + 6]) * qs;
            }
        } else {
            // y = [x0-x2+x3, x1-x2, x0, x1-x2+x3] per local group of 4
#pragma unroll
            for (int b = 0; b < EPT; b += 4) {
                f[b + 0] = (q[b + 0] - q[b + 2] + q[b + 3]) * qs;
                f[b + 1] = (q[b + 1] - q[b + 2]) * qs;
                f[b + 2] = q[b + 0] * qs;
                f[b + 3] = (q[b + 1] - q[b + 2] + q[b + 3]) * qs;
            }
        }
    }

    // ---- epilogue: x*scale + bias, nontemporal 128-bit stores ----
    const float* sp = scale + t * EPT;
    const float* bp = bias + t * EPT;
    float* op = out + base;
#pragma unroll
    for (int k = 0; k < 4; ++k) {
        v4f s4 = *(const v4f*)(sp + 4 * k);
        v4f b4 = *(const v4f*)(bp + 4 * k);
        v4f o;
        o[0] = f[4 * k + 0] * s4[0] + b4[0];
        o[1] = f[4 * k + 1] * s4[1] + b4[1];
        o[2] = f[4 * k + 2] * s4[2] + b4[2];
        o[3] = f[4 * k + 3] * s4[3] + b4[3];
        __builtin_nontemporal_store(o, (v4f*)(op + 4 * k));
    }
}

extern "C" void kernel_launch(void* const* d_in, const int* in_sizes, int n_in,
                              void* d_out, int out_size, void* d_ws, size_t ws_size,
                              hipStream_t stream) {
    (void)n_in; (void)out_size; (void)d_ws; (void)ws_size;
    const float* x     = (const float*)d_in[0];
    const float* scale = (const float*)d_in[1];
    const float* bias  = (const float*)d_in[2];
    float* out = (float*)d_out;

    const int rows = in_sizes[0] / DIM;   // 4*4096 = 16384 rows of length 4096
    twist_mj_rule42_kernel<<<dim3(rows), dim3(TPB), 0, stream>>>(x, scale, bias, out);
}